// GIN_40570261078219
// MI455X (gfx1250) — compile-verified
//
#include <hip/hip_runtime.h>

typedef float v2f __attribute__((ext_vector_type(2)));
typedef float v8f __attribute__((ext_vector_type(8)));

#define HDIM 64

// ---------------------------------------------------------------------------
// Utility kernels
// ---------------------------------------------------------------------------
__global__ void copy_f4(const float4* __restrict__ src, float4* __restrict__ dst, int n4) {
    int i = blockIdx.x * blockDim.x + threadIdx.x;
    if (i < n4) dst[i] = src[i];
}

__global__ void zero_f(float* __restrict__ p, int n) {
    int i = blockIdx.x * blockDim.x + threadIdx.x;
    if (i < n) p[i] = 0.0f;
}

// z[dst] += h[src] over all edges, D = feature width (multiple of 4).
// One thread handles one (edge, 4-feature) chunk: coalesced float4 gather,
// four global f32 atomic adds (L2-resident destinations).
template <int D>
__global__ void edge_scatter(const float* __restrict__ h, float* __restrict__ z,
                             const int* __restrict__ src, const int* __restrict__ dst,
                             int nedge) {
    constexpr int D4 = D / 4;
    long i = (long)blockIdx.x * blockDim.x + threadIdx.x;
    long total = (long)nedge * D4;
    if (i >= total) return;
    int e = (int)(i / D4);
    int f = (int)(i % D4) * 4;
    int s = src[e];
    int t = dst[e];
    const float4 v = *(const float4*)(h + (size_t)s * D + f);
    float* zp = z + (size_t)t * D + f;
    atomicAdd(zp + 0, v.x);
    atomicAdd(zp + 1, v.y);
    atomicAdd(zp + 2, v.z);
    atomicAdd(zp + 3, v.w);
}

// g[batch[n]] += h[n], 64-wide features.
__global__ void pool_scatter(const float* __restrict__ h, const int* __restrict__ batch,
                             float* __restrict__ g, int nnode) {
    int i = blockIdx.x * blockDim.x + threadIdx.x;
    int total = nnode * 16;  // 64/4 chunks
    if (i >= total) return;
    int node = i >> 4;
    int f = (i & 15) * 4;
    int b = batch[node];
    const float4 v = *(const float4*)(h + (size_t)node * 64 + f);
    float* gp = g + (size_t)b * 64 + f;
    atomicAdd(gp + 0, v.x);
    atomicAdd(gp + 1, v.y);
    atomicAdd(gp + 2, v.z);
    atomicAdd(gp + 3, v.w);
}

// ---------------------------------------------------------------------------
// Fused 2-layer MLP: OUT = act2( relu(Z @ W1 + b1) @ W2 + b2 )
//   Z: [nrows x DIN_T], W1: [DIN_T x 64], W2: [64 x NCOL2]
// Block = 128 threads (4 wave32), 16 rows of Z per block (rows % 16 == 0).
// Each wave owns a 16-wide output column tile; K-loop in steps of 4 using
// V_WMMA_F32_16X16X4_F32 (full f32 — matches the f32 reference numerics).
//
// f32 16x4 A layout : lanes 0-15 -> M=lane,   K = {k, k+1}
//                     lanes 16-31 -> M=lane-16, K = {k+2, k+3}
// f32 4x16 B layout : mirrored on K, N = lane&15
// f32 16x16 C layout: VGPR r, lanes 0-15 -> M=r, lanes 16-31 -> M=r+8, N=lane&15
// ---------------------------------------------------------------------------
template <int DIN_T, int NCOL2, bool RELU2>
__global__ __launch_bounds__(128) void gin_mlp_wmma(
    const float* __restrict__ Z,
    const float* __restrict__ W1, const float* __restrict__ B1,
    const float* __restrict__ W2, const float* __restrict__ B2,
    float* __restrict__ OUT) {
    constexpr int ZS = DIN_T + 2;  // padded LDS stride (even -> conflict-free float2)
    __shared__ __align__(16) float zt[16 * ZS];
    __shared__ __align__(16) float h1[16 * 66];

    const int tid   = threadIdx.x;
    const int wave  = tid >> 5;
    const int lane  = tid & 31;
    const int m     = lane & 15;          // M (A) / N (B,C) index within tile
    const int koff  = (lane >> 4) << 1;   // 0 or 2
    const int rbase = (lane >> 4) << 3;   // 0 or 8 (C-matrix row base)
    const long rowbase = (long)blockIdx.x * 16;

    // Stage Z tile (16 x DIN_T) into LDS as float2 with padded stride.
    constexpr int C2 = DIN_T / 2;
    for (int i = tid; i < 16 * C2; i += 128) {
        int r = i / C2;
        int c = (i % C2) * 2;
        v2f v = *(const v2f*)(Z + (rowbase + r) * DIN_T + c);
        *(v2f*)(zt + r * ZS + c) = v;
    }
    __syncthreads();

    // ---- Stage 1: H1 = relu(Z @ W1 + b1), wave handles cols [16*wave, +16) ----
    {
        const int cb = wave * 16;
        v8f acc = {};
        #pragma unroll
        for (int k = 0; k < DIN_T; k += 4) {
            v2f a = *(const v2f*)(zt + m * ZS + k + koff);
            v2f b;
            b.x = W1[(k + koff + 0) * HDIM + cb + m];
            b.y = W1[(k + koff + 1) * HDIM + cb + m];
            acc = __builtin_amdgcn_wmma_f32_16x16x4_f32(
                false, a, false, b, (short)0, acc, false, false);
        }
        const float bias = B1[cb + m];
        #pragma unroll
        for (int r = 0; r < 8; ++r) {
            float v = acc[r] + bias;
            h1[(rbase + r) * 66 + cb + m] = fmaxf(v, 0.0f);
        }
    }
    __syncthreads();

    // ---- Stage 2: OUT = act2(H1 @ W2 + b2) ----
    if (wave < NCOL2 / 16) {
        const int cb = wave * 16;
        v8f acc = {};
        #pragma unroll
        for (int k = 0; k < HDIM; k += 4) {
            v2f a = *(const v2f*)(h1 + m * 66 + k + koff);
            v2f b;
            b.x = W2[(k + koff + 0) * NCOL2 + cb + m];
            b.y = W2[(k + koff + 1) * NCOL2 + cb + m];
            acc = __builtin_amdgcn_wmma_f32_16x16x4_f32(
                false, a, false, b, (short)0, acc, false, false);
        }
        const float bias = B2[cb + m];
        #pragma unroll
        for (int r = 0; r < 8; ++r) {
            float v = acc[r] + bias;
            if (RELU2) v = fmaxf(v, 0.0f);
            OUT[(rowbase + rbase + r) * NCOL2 + cb + m] = v;
        }
    }
}

// ---------------------------------------------------------------------------
// Host orchestration (graph-capture safe: only kernel launches on `stream`)
// ---------------------------------------------------------------------------
extern "C" void kernel_launch(void* const* d_in, const int* in_sizes, int n_in,
                              void* d_out, int out_size, void* d_ws, size_t ws_size,
                              hipStream_t stream) {
    const float* x     = (const float*)d_in[0];
    const int*   eidx  = (const int*)d_in[1];
    const int*   batch = (const int*)d_in[2];
    const float* w1_0  = (const float*)d_in[3];
    const float* b1_0  = (const float*)d_in[4];
    const float* w2_0  = (const float*)d_in[5];
    const float* b2_0  = (const float*)d_in[6];
    const float* w1_r  = (const float*)d_in[7];
    const float* b1_r  = (const float*)d_in[8];
    const float* w2_r  = (const float*)d_in[9];
    const float* b2_r  = (const float*)d_in[10];
    const float* mw1   = (const float*)d_in[11];
    const float* mb1   = (const float*)d_in[12];
    const float* mw2   = (const float*)d_in[13];
    const float* mb2   = (const float*)d_in[14];
    float* out = (float*)d_out;

    const int N = in_sizes[0] / 128;
    const int E = in_sizes[1] / 2;
    const int G = out_size / 16;
    const int* src = eidx;
    const int* dst = eidx + E;

    // Workspace layout (floats): z0[N*128] | hA[N*64] | hB[N*64] | g[G*64]
    float* ws = (float*)d_ws;
    float* z0 = ws;
    float* hA = z0 + (size_t)N * 128;
    float* hB = hA + (size_t)N * 64;
    float* g  = hB + (size_t)N * 64;

    // ---- Layer 0: DIN=128 ----
    {
        int n4 = N * 128 / 4;
        copy_f4<<<(n4 + 255) / 256, 256, 0, stream>>>((const float4*)x, (float4*)z0, n4);
        long tot = (long)E * 32;
        edge_scatter<128><<<(int)((tot + 255) / 256), 256, 0, stream>>>(x, z0, src, dst, E);
        gin_mlp_wmma<128, 64, true><<<N / 16, 128, 0, stream>>>(z0, w1_0, b1_0, w2_0, b2_0, hA);
    }

    // ---- Layers 1..4: DIN=64 ----
    float* cur = hA;
    float* nxt = hB;
    for (int i = 0; i < 4; ++i) {
        int n4 = N * 64 / 4;
        copy_f4<<<(n4 + 255) / 256, 256, 0, stream>>>((const float4*)cur, (float4*)z0, n4);
        long tot = (long)E * 16;
        edge_scatter<64><<<(int)((tot + 255) / 256), 256, 0, stream>>>(cur, z0, src, dst, E);
        gin_mlp_wmma<64, 64, true><<<N / 16, 128, 0, stream>>>(
            z0, w1_r + (size_t)i * 64 * 64, b1_r + (size_t)i * 64,
            w2_r + (size_t)i * 64 * 64, b2_r + (size_t)i * 64, nxt);
        float* t = cur; cur = nxt; nxt = t;
    }

    // ---- Global add-pool + readout MLP ----
    {
        int ng = G * 64;
        zero_f<<<(ng + 255) / 256, 256, 0, stream>>>(g, ng);
        int tot = N * 16;
        pool_scatter<<<(tot + 255) / 256, 256, 0, stream>>>(cur, batch, g, N);
        gin_mlp_wmma<64, 16, false><<<G / 16, 128, 0, stream>>>(g, mw1, mb1, mw2, mb2, out);
    }
}